// QuPoolSX_46952582479903
// MI455X (gfx1250) — compile-verified
//
#include <hip/hip_runtime.h>

typedef float v2f __attribute__((ext_vector_type(2)));
typedef float v8f __attribute__((ext_vector_type(8)));

#define N_QUBITS 10
#define DIM      1024
#define BATCH    16
#define W_MUL    0.6324555320336759f   /* sqrt(2/5) */
#define LDPAD    33                    /* 32 + 1: kills LDS bank conflicts on row-strided WMMA feeds */
#define SADDR(i) ((((i) >> 5) * LDPAD) + ((i) & 31))

__global__ __launch_bounds__(128)
void qupool_sv_kernel(const float* __restrict__ x,      // (B,1,1024) complex64 interleaved
                      const float* __restrict__ weight, // 18 f32
                      float* __restrict__ out)          // (B,1,1024) complex64 interleaved
{
    __shared__ float sRe[32 * LDPAD];
    __shared__ float sIm[32 * LDPAD];

    const int b = blockIdx.x;
    const int t = threadIdx.x;              // 0..127, 4 waves of 32

    // ---- load |psi_b> into LDS (padded 32x33 layout, idx = i*32 + j) ----
    const float* xb = x + (size_t)b * (2 * DIM);
    for (int idx = t; idx < DIM; idx += 128) {
        float re = xb[2 * idx + 0];
        float im = xb[2 * idx + 1];
        int a = SADDR(idx);
        sRe[a] = re;
        sIm[a] = im;
    }
    __syncthreads();

    // ---- 9 fused layers: G = (I (x) RX(-w1)) * CNOT * RXX(w0) on qubit pair (q, q+1) ----
    // G(re,im):
    //   row0: (A,0) (0, B) (C,0) (0,-D)
    //   row1: (0,B) (A, 0) (0,-D) (C,0)
    //   row2: (0,-D) (C,0) (0, B) (A,0)
    //   row3: (C,0) (0,-D) (A,0) (0, B)
    // with A=c2*c, B=s2*c, C=s2*s, D=c2*s;  c,s = cos/sin(w0/2); c2,s2 = cos/sin(w1/2)
    for (int q = 0; q < N_QUBITS - 1; ++q) {
        float wa = weight[2 * q]     * W_MUL;
        float wb = weight[2 * q + 1] * W_MUL;
        float c  = __cosf(0.5f * wa), s  = __sinf(0.5f * wa);
        float c2 = __cosf(0.5f * wb), s2 = __sinf(0.5f * wb);
        float A = c2 * c, B = s2 * c, C = s2 * s, D = c2 * s;

        const int p  = (N_QUBITS - 1) - q;  // high bit of the adjacent pair
        const int pl = p - 1;               // low bit
        const int lowmask = (1 << pl) - 1;

        #pragma unroll
        for (int gi = 0; gi < 2; ++gi) {
            int g    = t + gi * 128;        // 256 disjoint groups of 4 amplitudes
            int base = ((g >> pl) << (p + 1)) | (g & lowmask);
            int i0 = SADDR(base);
            int i1 = SADDR(base | (1 << pl));
            int i2 = SADDR(base | (1 << p));
            int i3 = SADDR(base | (1 << p) | (1 << pl));

            float xr0 = sRe[i0], xi0 = sIm[i0];
            float xr1 = sRe[i1], xi1 = sIm[i1];
            float xr2 = sRe[i2], xi2 = sIm[i2];
            float xr3 = sRe[i3], xi3 = sIm[i3];

            float y0r =  A * xr0 - B * xi1 + C * xr2 + D * xi3;
            float y0i =  A * xi0 + B * xr1 + C * xi2 - D * xr3;
            float y1r = -B * xi0 + A * xr1 + D * xi2 + C * xr3;
            float y1i =  B * xr0 + A * xi1 - D * xr2 + C * xi3;
            float y2r =  D * xi0 + C * xr1 - B * xi2 + A * xr3;
            float y2i = -D * xr0 + C * xi1 + B * xr2 + A * xi3;
            float y3r =  C * xr0 + D * xi1 + A * xr2 - B * xi3;
            float y3i =  C * xi0 - D * xr1 + A * xi2 + B * xr3;

            sRe[i0] = y0r; sIm[i0] = y0i;
            sRe[i1] = y1r; sIm[i1] = y1i;
            sRe[i2] = y2r; sIm[i2] = y2i;
            sRe[i3] = y3r; sIm[i3] = y3i;
        }
        __syncthreads();
    }

    // ---- partial trace: out[j,k] = sum_i M[i,j] * conj(M[i,k]),  M = 32(traced) x 32(kept) ----
    // Real GEMMs via V_WMMA_F32_16X16X4_F32; contraction over rows => A (=M^T) and B (=M)
    // operands use the identical "row i, col = lane" LDS read pattern.
    const int lane = t & 31;
    const int wv   = t >> 5;                 // 4 waves -> four 16x16 tiles of the 32x32 output
    const int jt   = (wv >> 1) << 4;
    const int kt   = (wv & 1) << 4;
    const int colj = jt + (lane & 15);
    const int colk = kt + (lane & 15);
    const int rofs = (lane < 16) ? 0 : 2;    // half-wave K split per 32-bit A/B layout

    v8f accR = {0.f, 0.f, 0.f, 0.f, 0.f, 0.f, 0.f, 0.f};
    v8f accI = {0.f, 0.f, 0.f, 0.f, 0.f, 0.f, 0.f, 0.f};

    #pragma unroll
    for (int s8 = 0; s8 < 8; ++s8) {         // K = 32 rows, 4 per WMMA
        int r0 = 4 * s8 + rofs;
        int r1 = r0 + 1;
        v2f aR = { sRe[r0 * LDPAD + colj], sRe[r1 * LDPAD + colj] };
        v2f aI = { sIm[r0 * LDPAD + colj], sIm[r1 * LDPAD + colj] };
        v2f bR = { sRe[r0 * LDPAD + colk], sRe[r1 * LDPAD + colk] };
        v2f bI = { sIm[r0 * LDPAD + colk], sIm[r1 * LDPAD + colk] };
        v2f aRn = { -aR.x, -aR.y };          // f32 WMMA has no A/B NEG modifier (ISA NEG table)

        // re += aR*bR + aI*bI ; im += aI*bR - aR*bI
        accR = __builtin_amdgcn_wmma_f32_16x16x4_f32(false, aR,  false, bR, (short)0, accR, false, false);
        accR = __builtin_amdgcn_wmma_f32_16x16x4_f32(false, aI,  false, bI, (short)0, accR, false, false);
        accI = __builtin_amdgcn_wmma_f32_16x16x4_f32(false, aI,  false, bR, (short)0, accI, false, false);
        accI = __builtin_amdgcn_wmma_f32_16x16x4_f32(false, aRn, false, bI, (short)0, accI, false, false);
    }

    // ---- write D: 16x16 f32 tile layout, VGPR v -> M = v + (lane>=16 ? 8 : 0), N = lane&15 ----
    float* ob = out + (size_t)b * (2 * DIM);
    #pragma unroll
    for (int v = 0; v < 8; ++v) {
        int j = jt + v + ((lane < 16) ? 0 : 8);
        int k = kt + (lane & 15);
        int e = (j << 5) | k;
        ob[2 * e + 0] = accR[v];
        ob[2 * e + 1] = accI[v];
    }
}

extern "C" void kernel_launch(void* const* d_in, const int* in_sizes, int n_in,
                              void* d_out, int out_size, void* d_ws, size_t ws_size,
                              hipStream_t stream) {
    (void)in_sizes; (void)n_in; (void)d_ws; (void)ws_size; (void)out_size;
    const float* x = (const float*)d_in[0];   // complex64 interleaved (16,1,1024)
    const float* w = (const float*)d_in[1];   // f32[18]
    // d_in[2] = dimA (int, == 5) — fixed by N_QUBITS/2, not needed at runtime
    qupool_sv_kernel<<<BATCH, 128, 0, stream>>>(x, w, (float*)d_out);
}